// Linker_65592740544758
// MI455X (gfx1250) — compile-verified
//
#include <hip/hip_runtime.h>
#include <hip/hip_bf16.h>
#include <math.h>

typedef __attribute__((ext_vector_type(2))) float v2f;
typedef __attribute__((ext_vector_type(8))) float v8f;

#define B_ 8
#define S_ 4096
#define H_ 1024
#define L_ 32
#define M_ 4

// -------------------------------------------------------------------------
// Kernel 1: per-(b,l) span mean pooling + elementwise max over M spans.
// grid = B*L blocks, 256 threads; each thread owns 4 consecutive channels
// (one float4 / global_load_b128 per row, coalesced across the wave).
// seq_hiddens (134 MB) fits in the 192 MB L2, so this is latency-bound;
// prefetch the next span's row to overlap gathers.
// -------------------------------------------------------------------------
__global__ __launch_bounds__(256) void span_mean_max_kernel(
    const float* __restrict__ seq,   // [B,S,H]
    const int*   __restrict__ spans, // [B,L,M,2]
    float*       __restrict__ hid)   // [B*L, H]
{
    const int bl  = blockIdx.x;      // 0..B*L-1
    const int b   = bl / L_;
    const int tid = threadIdx.x;
    const int h0  = tid * 4;

    const int* sp = spans + bl * (M_ * 2);

    float bx = -INFINITY, by = -INFINITY, bz = -INFINITY, bw = -INFINITY;

    for (int m = 0; m < M_; ++m) {
        const int st = sp[m * 2 + 0];
        const int en = sp[m * 2 + 1];

        if (m + 1 < M_) {
            const int pst = sp[(m + 1) * 2 + 0];
            __builtin_prefetch(seq + ((size_t)b * S_ + pst) * H_ + h0, 0, 3);
        }

        float sx = 0.f, sy = 0.f, sz = 0.f, sw = 0.f;
        for (int r = st; r <= en; ++r) {
            const float4 v = *(const float4*)(seq + ((size_t)b * S_ + r) * H_ + h0);
            sx += v.x; sy += v.y; sz += v.z; sw += v.w;
        }
        const float inv = 1.0f / (float)(en - st + 1);
        sx *= inv; sy *= inv; sz *= inv; sw *= inv;

        bx = fmaxf(bx, sx);
        by = fmaxf(by, sy);
        bz = fmaxf(bz, sz);
        bw = fmaxf(bw, sw);
    }

    float4 out;
    out.x = bx; out.y = by; out.z = bz; out.w = bw;
    *(float4*)(hid + (size_t)bl * H_ + h0) = out;
}

// -------------------------------------------------------------------------
// Kernel 2: logits[256] = hiddens[256x1024] @ w[1024] + b via
// V_WMMA_F32_16X16X4_F32. One block of 16 waves; each wave owns a 16-row
// tile, K loop of 1024 in steps of 4 -> 256 wmma ops per wave.
// B operand replicates w across all 16 columns, so every D column holds the
// same logits; lanes with N==0 (lane 0 -> M=0..7, lane 16 -> M=8..15) write.
// A-layout (32-bit 16x4): lanes 0-15 hold K={0,1}, lanes 16-31 hold K={2,3}.
// -------------------------------------------------------------------------
__global__ __launch_bounds__(512) void gemv_wmma_kernel(
    const float* __restrict__ hid,    // [256, 1024]
    const float* __restrict__ w,      // [1024]
    const float* __restrict__ bias,   // [1]
    float*       __restrict__ logits) // [256]
{
    const int lane = threadIdx.x & 31;
    const int wave = threadIdx.x >> 5;          // 0..15 -> 16-row tile
    const int row  = wave * 16 + (lane & 15);   // 0..255
    const int koff = (lane >> 4) << 1;          // 0 (lanes 0-15) or 2 (lanes 16-31)

    const float* __restrict__ arow = hid + (size_t)row * H_;

    v8f c = {};
    #pragma unroll 8
    for (int k = 0; k < H_; k += 4) {
        v2f a  = *(const v2f*)(arow + k + koff);  // A: rows of hiddens
        v2f bb = *(const v2f*)(w    + k + koff);  // B: w replicated in N
        c = __builtin_amdgcn_wmma_f32_16x16x4_f32(
                /*neg_a=*/false, a, /*neg_b=*/false, bb,
                /*c_mod=*/(short)0, c, /*reuse_a=*/false, /*reuse_b=*/false);
    }

    const float bv = *bias;
    if ((lane & 15) == 0) {
        const int mbase = wave * 16 + ((lane >> 4) << 3); // lane0: M=0..7, lane16: M=8..15
        #pragma unroll
        for (int v = 0; v < 8; ++v) {
            logits[mbase + v] = c[v] + bv;
        }
    }
}

// -------------------------------------------------------------------------
// Kernel 3: per-batch argmax over L=32 logits. L == wave32 exactly: one wave
// per batch, shuffle-based max/argmax with first-index tie-break (matches
// jnp.argmax). best_idx written as float (d_out is a float buffer).
// -------------------------------------------------------------------------
__global__ __launch_bounds__(256) void argmax_kernel(
    const float* __restrict__ logits, // [B, L]
    float*       __restrict__ best)   // [B]
{
    const int wave = threadIdx.x >> 5; // batch 0..7
    const int lane = threadIdx.x & 31; // link index 0..31

    float v   = logits[wave * L_ + lane];
    int   idx = lane;

    #pragma unroll
    for (int off = 16; off > 0; off >>= 1) {
        const float ov = __shfl_xor(v, off, 32);
        const int   oi = __shfl_xor(idx, off, 32);
        if (ov > v || (ov == v && oi < idx)) { v = ov; idx = oi; }
    }

    if (lane == 0) best[wave] = (float)idx;
}

extern "C" void kernel_launch(void* const* d_in, const int* in_sizes, int n_in,
                              void* d_out, int out_size, void* d_ws, size_t ws_size,
                              hipStream_t stream) {
    (void)in_sizes; (void)n_in; (void)out_size; (void)d_ws; (void)ws_size;

    const float* seq   = (const float*)d_in[0]; // [B,S,H]
    const int*   spans = (const int*)  d_in[1]; // [B,L,M,2]
    const float* w     = (const float*)d_in[2]; // [H]
    const float* bias  = (const float*)d_in[3]; // [1]

    float* logits = (float*)d_out;                       // [B*L]     = 256
    float* hid    = logits + (B_ * L_);                  // [B*L, H]  = 262144
    float* best   = hid + (size_t)B_ * L_ * H_;          // [B]       = 8

    span_mean_max_kernel<<<B_ * L_, 256, 0, stream>>>(seq, spans, hid);
    gemv_wmma_kernel<<<1, 512, 0, stream>>>(hid, w, bias, logits);
    argmax_kernel<<<1, 256, 0, stream>>>(logits, best);
}